// GATedge_82136954568958
// MI455X (gfx1250) — compile-verified
//
#include <hip/hip_runtime.h>
#include <math.h>

// Problem constants (from reference)
#define BB 16
#define JJ 512
#define KK 64
#define INF_ 64   // IN
#define HD 64
#define NEG_SLOPE 0.2f

typedef __attribute__((ext_vector_type(2))) float v2f;
typedef __attribute__((ext_vector_type(8))) float v8f;

// ---------------------------------------------------------------------------
// helpers
// ---------------------------------------------------------------------------
__device__ __forceinline__ float leaky(float x) {
    return x >= 0.0f ? x : NEG_SLOPE * x;
}
__device__ __forceinline__ float sigmoidf(float x) {
    return 1.0f / (1.0f + __expf(-x));
}
__device__ __forceinline__ float wave_max(float v) {
    #pragma unroll
    for (int o = 16; o > 0; o >>= 1) v = fmaxf(v, __shfl_xor(v, o, 32));
    return v;
}
__device__ __forceinline__ float wave_sum(float v) {
    #pragma unroll
    for (int o = 16; o > 0; o >>= 1) v += __shfl_xor(v, o, 32);
    return v;
}

// ---------------------------------------------------------------------------
// Generic fp32 WMMA GEMM: C[b] = Aeff[b] (MxKin) @ B[b] (KinxN)
// TRANSA=0: Aeff[m][k] = A[m*lda + k]
// TRANSA=1: Aeff[m][k] = A[k*lda + m]   (alpha matrices stored softmax-major)
// One wave (32 threads) computes one 16x16 tile via V_WMMA_F32_16X16X4_F32.
// M,N multiples of 16; Kin multiple of 16. EXEC all-ones (no divergence).
//
// ISA VGPR layouts (CDNA5 ISA):
//   A 16x4 f32: lane<16 -> row=lane, K=k0,k0+1 ; lane>=16 -> row=lane-16, K=k0+2,k0+3
//   B 4x16    : vgpr v, lanes0-15 -> K=k0+v? mirrored: lane<16 K=k0/k0+1, lane>=16 K=k0+2/k0+3
//   C/D 16x16 : vgpr r: row = half*8 + r, col = lane&15
// ---------------------------------------------------------------------------
template <bool TRANSA>
__global__ __launch_bounds__(32) void wmma_gemm_f32(
    const float* __restrict__ A, const float* __restrict__ B, float* __restrict__ C,
    int M, int N, int Kin, int lda, int ldb, int ldc,
    long long strideA, long long strideB, long long strideC)
{
    const int lane  = threadIdx.x;
    const int half  = lane >> 4;   // 0: lanes 0-15, 1: lanes 16-31
    const int l16   = lane & 15;
    const int tilesN = N >> 4;
    const int tm = (blockIdx.x / tilesN) << 4;
    const int tn = (blockIdx.x % tilesN) << 4;

    const float* Ab = A + (long long)blockIdx.y * strideA;
    const float* Bb = B + (long long)blockIdx.y * strideB;
    float*       Cb = C + (long long)blockIdx.y * strideC;

    v8f acc = {0.f, 0.f, 0.f, 0.f, 0.f, 0.f, 0.f, 0.f};

    #pragma unroll 4
    for (int k0 = 0; k0 < Kin; k0 += 4) {
        const int ka = k0 + (half << 1);
        v2f a, b;
        if (TRANSA) {
            a.x = Ab[(long long)ka * lda + tm + l16];
            a.y = Ab[(long long)(ka + 1) * lda + tm + l16];
        } else {
            a.x = Ab[(long long)(tm + l16) * lda + ka];
            a.y = Ab[(long long)(tm + l16) * lda + ka + 1];
        }
        b.x = Bb[(long long)ka * ldb + tn + l16];
        b.y = Bb[(long long)(ka + 1) * ldb + tn + l16];
        acc = __builtin_amdgcn_wmma_f32_16x16x4_f32(
            /*neg_a=*/false, a, /*neg_b=*/false, b,
            /*c_mod=*/(short)0, acc, /*reuse_a=*/false, /*reuse_b=*/false);
    }

    const int row0 = tm + (half << 3);
    #pragma unroll
    for (int r = 0; r < 8; ++r)
        Cb[(long long)(row0 + r) * ldc + tn + l16] = acc[r];
}

// ---------------------------------------------------------------------------
// c_e = dot(w_edge, attn_l)
// ---------------------------------------------------------------------------
__global__ void k_dot_ce(const float* __restrict__ w_edge,
                         const float* __restrict__ attn_l,
                         float* __restrict__ ce)
{
    __shared__ float sh[64];
    int t = threadIdx.x;
    sh[t] = w_edge[t] * attn_l[t];
    __syncthreads();
    for (int s = 32; s > 0; s >>= 1) {
        if (t < s) sh[t] += sh[t + s];
        __syncthreads();
    }
    if (t == 0) ce[0] = sh[0];
}

// out[row] = dot(in[row, 0:64], vec)   (el / er)
__global__ void k_rowdot(const float* __restrict__ in, const float* __restrict__ vec,
                         float* __restrict__ out, int nrows)
{
    int r = blockIdx.x * blockDim.x + threadIdx.x;
    if (r >= nrows) return;
    const float* p = in + (long long)r * HD;
    float acc = 0.f;
    #pragma unroll
    for (int d = 0; d < HD; ++d) acc += p[d] * vec[d];
    out[r] = acc;
}

// eijk = leaky(adj_om * (el + er) + edge_raw * c_e)
__global__ void k_eijk(const float* __restrict__ station, const int* __restrict__ bidx,
                       const float* __restrict__ el, const float* __restrict__ er,
                       const float* __restrict__ edge_raw, const float* __restrict__ ce,
                       float* __restrict__ eijk)
{
    int idx = blockIdx.x * blockDim.x + threadIdx.x;
    if (idx >= BB * JJ * KK) return;
    int k = idx & (KK - 1);
    int j = (idx >> 6) & (JJ - 1);
    int b = idx >> 15;
    int bb = bidx[b];
    float adj = station[((long long)bb * JJ + j) * KK + k];
    float e = adj * (el[b * JJ + j] + er[b * KK + k]) + edge_raw[idx] * ce[0];
    eijk[idx] = leaky(e);
}

// softmax over axis1 (j in 0..J-1 masked by adj, plus self row ekk) per (b,k)
// one wave per (b,k); also fuses s1 = sum_j alpha * edge_raw
__global__ __launch_bounds__(32) void k_softmax_om(
    const float* __restrict__ station, const int* __restrict__ bidx,
    const float* __restrict__ eijk, const float* __restrict__ er,
    const float* __restrict__ edge_raw,
    float* __restrict__ alpha_ijk, float* __restrict__ alpha_kk,
    float* __restrict__ s1)
{
    int bk = blockIdx.x;           // b*KK + k
    int b  = bk >> 6;
    int k  = bk & (KK - 1);
    int bb = bidx[b];
    int lane = threadIdx.x;

    float extra = leaky(2.0f * er[bk]);

    float m = -INFINITY;
    for (int j = lane; j < JJ; j += 32) {
        long long ai = ((long long)bb * JJ + j) * KK + k;
        long long ei = ((long long)b * JJ + j) * KK + k;
        if (station[ai] == 1.0f) m = fmaxf(m, eijk[ei]);
    }
    m = fmaxf(wave_max(m), extra);

    float se = 0.f, s1l = 0.f;
    for (int j = lane; j < JJ; j += 32) {
        long long ai = ((long long)bb * JJ + j) * KK + k;
        long long ei = ((long long)b * JJ + j) * KK + k;
        float a = 0.f;
        if (station[ai] == 1.0f) a = __expf(eijk[ei] - m);
        alpha_ijk[ei] = a;                 // un-normalized for now
        se  += a;
        s1l += a * edge_raw[ei];
    }
    se = wave_sum(se) + __expf(extra - m);
    float inv = 1.0f / se;
    // normalize
    for (int j = lane; j < JJ; j += 32) {
        long long ei = ((long long)b * JJ + j) * KK + k;
        alpha_ijk[ei] *= inv;
    }
    s1l = wave_sum(s1l);
    if (lane == 0) {
        alpha_kk[bk] = __expf(extra - m) * inv;
        s1[bk] = s1l * inv;
    }
}

// softmax over axis1 (k in 0..K-1 masked by adj, plus self row ell) per (b,j)
// writes alpha_kij[b,k,j]; fuses s2 = sum_k alpha * edge_raw
__global__ __launch_bounds__(32) void k_softmax_mo(
    const float* __restrict__ station, const int* __restrict__ bidx,
    const float* __restrict__ eijk, const float* __restrict__ el,
    const float* __restrict__ edge_raw,
    float* __restrict__ alpha_kij, float* __restrict__ alpha_ll1,
    float* __restrict__ s2)
{
    int bj = blockIdx.x;           // b*JJ + j
    int b  = bj >> 9;
    int j  = bj & (JJ - 1);
    int bb = bidx[b];
    int lane = threadIdx.x;

    float extra = leaky(2.0f * el[bj]);

    float v0 = -INFINITY, v1 = -INFINITY;
    int k0 = lane, k1 = lane + 32;
    long long arow = ((long long)bb * JJ + j) * KK;
    long long erow = ((long long)b * JJ + j) * KK;
    bool m0 = station[arow + k0] == 1.0f;
    bool m1 = station[arow + k1] == 1.0f;
    if (m0) v0 = eijk[erow + k0];
    if (m1) v1 = eijk[erow + k1];
    float m = fmaxf(wave_max(fmaxf(v0, v1)), extra);

    float a0 = m0 ? __expf(v0 - m) : 0.f;
    float a1 = m1 ? __expf(v1 - m) : 0.f;
    float se = wave_sum(a0 + a1) + __expf(extra - m);
    float inv = 1.0f / se;
    a0 *= inv; a1 *= inv;

    alpha_kij[((long long)b * KK + k0) * JJ + j] = a0;
    alpha_kij[((long long)b * KK + k1) * JJ + j] = a1;
    float s2l = wave_sum(a0 * edge_raw[erow + k0] + a1 * edge_raw[erow + k1]);
    if (lane == 0) {
        alpha_ll1[bj] = __expf(extra - m) * inv;
        s2[bj] = s2l;
    }
}

// softmax over axis1 (j masked by adj_oo, value = ell[b,j]; plus self ell[b,l]) per (b,l)
__global__ __launch_bounds__(32) void k_softmax_oo(
    const float* __restrict__ pre, const int* __restrict__ bidx,
    const float* __restrict__ el,
    float* __restrict__ alpha_jl, float* __restrict__ alpha_ll2)
{
    int bl = blockIdx.x;           // b*JJ + l
    int b  = bl >> 9;
    int l  = bl & (JJ - 1);
    int bb = bidx[b];
    int lane = threadIdx.x;

    float extra = leaky(2.0f * el[bl]);

    float m = -INFINITY;
    for (int j = lane; j < JJ; j += 32) {
        if (pre[((long long)bb * JJ + j) * JJ + l] == 1.0f)
            m = fmaxf(m, leaky(2.0f * el[b * JJ + j]));
    }
    m = fmaxf(wave_max(m), extra);

    float se = 0.f;
    for (int j = lane; j < JJ; j += 32) {
        float a = 0.f;
        if (pre[((long long)bb * JJ + j) * JJ + l] == 1.0f)
            a = __expf(leaky(2.0f * el[b * JJ + j]) - m);
        alpha_jl[((long long)b * JJ + j) * JJ + l] = a;
        se += a;
    }
    se = wave_sum(se) + __expf(extra - m);
    float inv = 1.0f / se;
    for (int j = lane; j < JJ; j += 32)
        alpha_jl[((long long)b * JJ + j) * JJ + l] *= inv;
    if (lane == 0) alpha_ll2[bl] = __expf(extra - m) * inv;
}

// nu_k_prime = sigmoid(bkd + s1*w_edge + feat_dst*alpha_kk)  -> d_out[0:B*K*HD]
__global__ void k_nu_out(const float* __restrict__ bkd, const float* __restrict__ s1,
                         const float* __restrict__ w_edge, const float* __restrict__ feat_dst,
                         const float* __restrict__ alpha_kk, float* __restrict__ out)
{
    int idx = blockIdx.x * blockDim.x + threadIdx.x;
    if (idx >= BB * KK * HD) return;
    int d  = idx & (HD - 1);
    int bk = idx >> 6;
    out[idx] = sigmoidf(bkd[idx] + s1[bk] * w_edge[d] + feat_dst[idx] * alpha_kk[bk]);
}

// mu1 = sigmoid(mu1 + s2*w_edge + feat_src*alpha_ll1)   (in place)
__global__ void k_mu1_ep(float* __restrict__ mu1, const float* __restrict__ s2,
                         const float* __restrict__ w_edge, const float* __restrict__ feat_src,
                         const float* __restrict__ alpha_ll1)
{
    int idx = blockIdx.x * blockDim.x + threadIdx.x;
    if (idx >= BB * JJ * HD) return;
    int d  = idx & (HD - 1);
    int bj = idx >> 6;
    mu1[idx] = sigmoidf(mu1[idx] + s2[bj] * w_edge[d] + feat_src[idx] * alpha_ll1[bj]);
}

// mu2 = sigmoid(mu2 + feat_src*alpha_ll2)   (in place)
__global__ void k_mu2_ep(float* __restrict__ mu2, const float* __restrict__ feat_src,
                         const float* __restrict__ alpha_ll2)
{
    int idx = blockIdx.x * blockDim.x + threadIdx.x;
    if (idx >= BB * JJ * HD) return;
    int bl = idx >> 6;
    mu2[idx] = sigmoidf(mu2[idx] + feat_src[idx] * alpha_ll2[bl]);
}

// e[b] = mean_j dot(tanh(t[b,j,:] + b_sem), attn_sem)
__global__ void k_sem_reduce(const float* __restrict__ t, const float* __restrict__ b_sem,
                             const float* __restrict__ attn_sem, float* __restrict__ e)
{
    __shared__ float sh[256];
    int b = blockIdx.x;
    int tid = threadIdx.x;
    float acc = 0.f;
    for (int j = tid; j < JJ; j += 256) {
        const float* row = t + ((long long)b * JJ + j) * HD;
        float s = 0.f;
        #pragma unroll
        for (int d = 0; d < HD; ++d) s += tanhf(row[d] + b_sem[d]) * attn_sem[d];
        acc += s;
    }
    sh[tid] = acc;
    __syncthreads();
    for (int s = 128; s > 0; s >>= 1) {
        if (tid < s) sh[tid] += sh[tid + s];
        __syncthreads();
    }
    if (tid == 0) e[b] = sh[0] * (1.0f / JJ);
}

// mu_ij_prime = w0*mu1 + w1*mu2, w = softmax([e1,e2]) -> d_out[B*K*HD : ]
__global__ void k_final(const float* __restrict__ mu1, const float* __restrict__ mu2,
                        const float* __restrict__ e1, const float* __restrict__ e2,
                        float* __restrict__ out)
{
    int idx = blockIdx.x * blockDim.x + threadIdx.x;
    if (idx >= BB * JJ * HD) return;
    int b = idx >> 15;                      // J*HD = 32768
    float w0 = sigmoidf(e1[b] - e2[b]);     // softmax([e1,e2])[0]
    out[idx] = w0 * mu1[idx] + (1.0f - w0) * mu2[idx];
}

// ---------------------------------------------------------------------------
extern "C" void kernel_launch(void* const* d_in, const int* in_sizes, int n_in,
                              void* d_out, int out_size, void* d_ws, size_t ws_size,
                              hipStream_t stream)
{
    const float* station  = (const float*)d_in[0];   // (B,J,K)
    const float* pre      = (const float*)d_in[1];   // (B,J,J)
    const int*   bidx     = (const int*)  d_in[2];   // (B,)
    const float* feat_opr = (const float*)d_in[3];   // (B,J,IN)
    const float* feat_mch = (const float*)d_in[4];   // (B,K,IN)
    const float* edge_raw = (const float*)d_in[5];   // (B,J,K)
    const float* W_src    = (const float*)d_in[6];   // (IN,HD)
    const float* W_dst    = (const float*)d_in[7];   // (IN,HD)
    const float* w_edge   = (const float*)d_in[8];   // (HD,)
    const float* attn_l   = (const float*)d_in[9];   // (HD,)
    const float* attn_r   = (const float*)d_in[10];  // (HD,)
    const float* attn_sem = (const float*)d_in[11];  // (HD,)
    const float* W_sem    = (const float*)d_in[12];  // (HD,HD)
    const float* b_sem    = (const float*)d_in[13];  // (HD,)

    float* out = (float*)d_out;          // [B*K*HD nu | B*J*HD mu]

    // workspace carve-up (floats)
    float* ws = (float*)d_ws;
    size_t o = 0;
    auto alloc = [&](size_t n) { float* p = ws + o; o += n; return p; };
    float* feat_src  = alloc((size_t)BB * JJ * HD);   // 524288
    float* feat_dst  = alloc((size_t)BB * KK * HD);   // 65536
    float* el        = alloc((size_t)BB * JJ);
    float* er        = alloc((size_t)BB * KK);
    float* eijk      = alloc((size_t)BB * JJ * KK);   // reused as t1 later
    float* alpha_ijk = alloc((size_t)BB * JJ * KK);   // reused as t2 later
    float* alpha_kk  = alloc((size_t)BB * KK);
    float* alpha_kij = alloc((size_t)BB * KK * JJ);
    float* alpha_jl  = alloc((size_t)BB * JJ * JJ);   // 4194304
    float* alpha_ll1 = alloc((size_t)BB * JJ);
    float* alpha_ll2 = alloc((size_t)BB * JJ);
    float* s1        = alloc((size_t)BB * KK);
    float* s2        = alloc((size_t)BB * JJ);
    float* bkd       = alloc((size_t)BB * KK * HD);
    float* mu1       = alloc((size_t)BB * JJ * HD);
    float* mu2       = alloc((size_t)BB * JJ * HD);
    float* e1        = alloc(BB);
    float* e2        = alloc(BB);
    float* ce        = alloc(1);
    (void)ws_size; (void)n_in; (void)in_sizes; (void)out_size;

    // 1) c_e = dot(w_edge, attn_l)
    k_dot_ce<<<1, 64, 0, stream>>>(w_edge, attn_l, ce);

    // 2) feat_src = feat_opr @ W_src   (M=B*J, N=HD, K=IN)
    wmma_gemm_f32<false><<<dim3((BB * JJ / 16) * (HD / 16), 1), 32, 0, stream>>>(
        feat_opr, W_src, feat_src, BB * JJ, HD, INF_, INF_, HD, HD, 0, 0, 0);
    // 3) feat_dst = feat_mch @ W_dst   (M=B*K)
    wmma_gemm_f32<false><<<dim3((BB * KK / 16) * (HD / 16), 1), 32, 0, stream>>>(
        feat_mch, W_dst, feat_dst, BB * KK, HD, INF_, INF_, HD, HD, 0, 0, 0);

    // 4) el, er
    k_rowdot<<<(BB * JJ + 255) / 256, 256, 0, stream>>>(feat_src, attn_l, el, BB * JJ);
    k_rowdot<<<(BB * KK + 255) / 256, 256, 0, stream>>>(feat_dst, attn_r, er, BB * KK);

    // 5) eijk
    k_eijk<<<(BB * JJ * KK + 255) / 256, 256, 0, stream>>>(
        station, bidx, el, er, edge_raw, ce, eijk);

    // 6) softmaxes (one wave per column)
    k_softmax_om<<<BB * KK, 32, 0, stream>>>(station, bidx, eijk, er, edge_raw,
                                             alpha_ijk, alpha_kk, s1);
    k_softmax_mo<<<BB * JJ, 32, 0, stream>>>(station, bidx, eijk, el, edge_raw,
                                             alpha_kij, alpha_ll1, s2);
    k_softmax_oo<<<BB * JJ, 32, 0, stream>>>(pre, bidx, el, alpha_jl, alpha_ll2);

    // 7) bkd[b] = alpha_ijk[b]^T (KxJ) @ feat_src[b] (JxHD)
    wmma_gemm_f32<true><<<dim3((KK / 16) * (HD / 16), BB), 32, 0, stream>>>(
        alpha_ijk, feat_src, bkd, KK, HD, JJ, KK, HD, HD,
        (long long)JJ * KK, (long long)JJ * HD, (long long)KK * HD);
    k_nu_out<<<(BB * KK * HD + 255) / 256, 256, 0, stream>>>(
        bkd, s1, w_edge, feat_dst, alpha_kk, out);

    // 8) mu1[b] = alpha_kij[b]^T (JxK) @ feat_dst[b] (KxHD) ; then epilogue in place
    wmma_gemm_f32<true><<<dim3((JJ / 16) * (HD / 16), BB), 32, 0, stream>>>(
        alpha_kij, feat_dst, mu1, JJ, HD, KK, JJ, HD, HD,
        (long long)KK * JJ, (long long)KK * HD, (long long)JJ * HD);
    k_mu1_ep<<<(BB * JJ * HD + 255) / 256, 256, 0, stream>>>(
        mu1, s2, w_edge, feat_src, alpha_ll1);

    // 9) mu2[b] = alpha_jl[b]^T (JxJ -> LxJ) @ feat_src[b] (JxHD) ; epilogue
    wmma_gemm_f32<true><<<dim3((JJ / 16) * (HD / 16), BB), 32, 0, stream>>>(
        alpha_jl, feat_src, mu2, JJ, HD, JJ, JJ, HD, HD,
        (long long)JJ * JJ, (long long)JJ * HD, (long long)JJ * HD);
    k_mu2_ep<<<(BB * JJ * HD + 255) / 256, 256, 0, stream>>>(mu2, feat_src, alpha_ll2);

    // 10) semantic attention: t1 = mu1 @ W_sem (reuse eijk), t2 = mu2 @ W_sem (reuse alpha_ijk)
    float* t1 = eijk;
    float* t2 = alpha_ijk;
    wmma_gemm_f32<false><<<dim3((BB * JJ / 16) * (HD / 16), 1), 32, 0, stream>>>(
        mu1, W_sem, t1, BB * JJ, HD, HD, HD, HD, HD, 0, 0, 0);
    k_sem_reduce<<<BB, 256, 0, stream>>>(t1, b_sem, attn_sem, e1);
    wmma_gemm_f32<false><<<dim3((BB * JJ / 16) * (HD / 16), 1), 32, 0, stream>>>(
        mu2, W_sem, t2, BB * JJ, HD, HD, HD, HD, HD, 0, 0, 0);
    k_sem_reduce<<<BB, 256, 0, stream>>>(t2, b_sem, attn_sem, e2);

    // 11) blend -> mu_ij_prime
    k_final<<<(BB * JJ * HD + 255) / 256, 256, 0, stream>>>(
        mu1, mu2, e1, e2, out + (size_t)BB * KK * HD);
}